// SAGE_19000935317832
// MI455X (gfx1250) — compile-verified
//
#include <hip/hip_runtime.h>

typedef float v2f __attribute__((ext_vector_type(2)));
typedef float v8f __attribute__((ext_vector_type(8)));

#define N_NODES 100000

// ---------------------------------------------------------------------------
// Fused SAGE projection GEMM:  Y[N, OUT] = X[N, IN] @ [Wself.T | Wneigh.T]
// Columns [0, HALF) get +bias (self part). fp32 WMMA 16x16x4, wave32.
// One wave computes a 16-row tile across all OUT/16 column tiles.
// ---------------------------------------------------------------------------
template <int IN, int OUT, int HALF>
__global__ void sage_gemm_wmma(const float* __restrict__ X,
                               const float* __restrict__ Wself,
                               const float* __restrict__ Wneigh,
                               const float* __restrict__ bias,
                               float* __restrict__ Y,
                               int nRowTiles) {
    __shared__ float Bs[IN * OUT];  // B matrix staged [k][n] for coalesced frags

    const int tid = threadIdx.x;
    for (int i = tid; i < IN * OUT; i += blockDim.x) {
        const int k = i / OUT;
        const int n = i % OUT;
        Bs[i] = (n < HALF) ? Wself[n * IN + k] : Wneigh[(n - HALF) * IN + k];
    }
    __syncthreads();

    const int wave = tid >> 5;
    const int lane = tid & 31;
    const int rowTile = blockIdx.x * (blockDim.x >> 5) + wave;
    if (rowTile >= nRowTiles) return;  // wave-uniform: EXEC stays all-ones

    const int rowBase = rowTile * 16;
    const int mrow = rowBase + (lane & 15);      // A-matrix row for this lane
    const int koff = (lane >> 4) << 1;           // lanes 0-15: K+{0,1}; 16-31: K+{2,3}
    const int ncol = lane & 15;                  // C/D column within tile
    const int hi = (lane >> 4) * 8;              // C/D row-half offset

    const float* xrow = X + (size_t)mrow * IN;

#pragma unroll
    for (int ct = 0; ct < OUT / 16; ++ct) {
        const int n = ct * 16 + ncol;
        v8f acc = {0.f, 0.f, 0.f, 0.f, 0.f, 0.f, 0.f, 0.f};
#pragma unroll
        for (int ks = 0; ks < IN / 4; ++ks) {
            const int k = ks * 4 + koff;
            const v2f a = *(const v2f*)(xrow + k);          // A frag: 2 consecutive K
            v2f b;
            b.x = Bs[k * OUT + n];                          // B frag VGPR0: row k
            b.y = Bs[(k + 1) * OUT + n];                    // B frag VGPR1: row k+1
            acc = __builtin_amdgcn_wmma_f32_16x16x4_f32(
                /*neg_a=*/false, a, /*neg_b=*/false, b,
                /*c_mod=*/(short)0, acc, /*reuse_a=*/false, /*reuse_b=*/false);
        }
        const float bv = (n < HALF) ? bias[n] : 0.0f;
#pragma unroll
        for (int r = 0; r < 8; ++r) {
            Y[(size_t)(rowBase + hi + r) * OUT + n] = acc[r] + bv;
        }
    }
}

// ---------------------------------------------------------------------------
// Helpers: zero, degree count, edge scatter-add (projected features), combine
// ---------------------------------------------------------------------------
__global__ void zero_f32(float* __restrict__ p, long n) {
    long i = (long)blockIdx.x * blockDim.x + threadIdx.x;
    if (i < n) p[i] = 0.0f;
}

__global__ void degree_kernel(const int* __restrict__ dst,
                              float* __restrict__ deg, int E) {
    int e = blockIdx.x * blockDim.x + threadIdx.x;
    if (e < E) atomicAdd(&deg[dst[e]], 1.0f);
}

// agg[dst] += Y[src*STRIDE + OFF .. +F]   (projected-neighbor scatter)
template <int F, int OFF, int STRIDE>
__global__ void scatter_add(const float* __restrict__ Y,
                            const int* __restrict__ src,
                            const int* __restrict__ dst,
                            float* __restrict__ agg, int E) {
    const int groups = F / 4;
    long tid = (long)blockIdx.x * blockDim.x + threadIdx.x;
    if (tid >= (long)E * groups) return;
    const int e = (int)(tid / groups);
    const int g = (int)(tid % groups) * 4;
    const int s = src[e];
    const int d = dst[e];
    const float4 v = *(const float4*)(Y + (size_t)s * STRIDE + OFF + g);
    float* base = agg + (size_t)d * F + g;
    atomicAdd(base + 0, v.x);
    atomicAdd(base + 1, v.y);
    atomicAdd(base + 2, v.z);
    atomicAdd(base + 3, v.w);
}

// H[v,f] = Y[v*STRIDE + f] (self proj, bias already in) + agg[v,f]/max(deg,1)
template <int F, int STRIDE>
__global__ void combine(const float* __restrict__ Y,
                        const float* __restrict__ agg,
                        const float* __restrict__ deg,
                        float* __restrict__ H, int Nn) {
    long tid = (long)blockIdx.x * blockDim.x + threadIdx.x;
    if (tid >= (long)Nn * F) return;
    const int v = (int)(tid / F);
    const int f = (int)(tid % F);
    const float inv = 1.0f / fmaxf(deg[v], 1.0f);
    H[tid] = Y[(size_t)v * STRIDE + f] + agg[tid] * inv;
}

// ---------------------------------------------------------------------------
extern "C" void kernel_launch(void* const* d_in, const int* in_sizes, int n_in,
                              void* d_out, int out_size, void* d_ws, size_t ws_size,
                              hipStream_t stream) {
    const float* x       = (const float*)d_in[0];  // [N,64]
    const int*   src     = (const int*)d_in[1];    // [E]
    const int*   dst     = (const int*)d_in[2];    // [E]
    const float* W1_self = (const float*)d_in[3];  // [32,64]
    const float* W1_nei  = (const float*)d_in[4];  // [32,64]
    const float* b1      = (const float*)d_in[5];  // [32]
    const float* W2_self = (const float*)d_in[6];  // [16,32]
    const float* W2_nei  = (const float*)d_in[7];  // [16,32]
    const float* b2      = (const float*)d_in[8];  // [16]
    float* out = (float*)d_out;                    // [N,16]

    const int N = N_NODES;
    const int E = in_sizes[1];

    // Workspace carve-up (floats). agg1|agg2|deg kept contiguous for one zero pass.
    float* ws   = (float*)d_ws;
    float* y1   = ws;                 // [N,64]  = [self32 | neigh32] (b1 in self)
    float* agg1 = y1 + (size_t)N * 64;   // [N,32]
    float* agg2 = agg1 + (size_t)N * 32; // [N,16]
    float* deg  = agg2 + (size_t)N * 16; // [N]
    float* h1   = deg + N;               // [N,32]
    float* z2   = h1 + (size_t)N * 32;   // [N,32] = [self16 | neigh16] (b2 in self)

    const int B = 256;

    // 1) zero accumulators (agg1 + agg2 + deg contiguous: N*49 floats)
    {
        long n = (long)N * 49;
        zero_f32<<<(unsigned)((n + B - 1) / B), B, 0, stream>>>(agg1, n);
    }
    // 2) in-degrees
    degree_kernel<<<(E + B - 1) / B, B, 0, stream>>>(dst, deg, E);

    // 3) layer-1 fused projection (WMMA): y1 = x @ [W1_self.T | W1_neigh.T] (+b1 on self)
    {
        const int rowTiles = N / 16;                   // 6250 exactly
        const int wavesPerBlk = B / 32;                // 8
        const int blocks = (rowTiles + wavesPerBlk - 1) / wavesPerBlk;
        sage_gemm_wmma<64, 64, 32><<<blocks, B, 0, stream>>>(
            x, W1_self, W1_nei, b1, y1, rowTiles);
    }
    // 4) scatter projected neighbor features (32 per edge, halved vs naive 64)
    {
        long items = (long)E * (32 / 4);
        scatter_add<32, 32, 64><<<(unsigned)((items + B - 1) / B), B, 0, stream>>>(
            y1, src, dst, agg1, E);
    }
    // 5) h1 = self + mean-neigh
    {
        long n = (long)N * 32;
        combine<32, 64><<<(unsigned)((n + B - 1) / B), B, 0, stream>>>(
            y1, agg1, deg, h1, N);
    }
    // 6) layer-2 fused projection (WMMA): z2 = h1 @ [W2_self.T | W2_neigh.T] (+b2 on self)
    {
        const int rowTiles = N / 16;
        const int wavesPerBlk = B / 32;
        const int blocks = (rowTiles + wavesPerBlk - 1) / wavesPerBlk;
        sage_gemm_wmma<32, 32, 16><<<blocks, B, 0, stream>>>(
            h1, W2_self, W2_nei, b2, z2, rowTiles);
    }
    // 7) scatter projected neighbor features (16 per edge)
    {
        long items = (long)E * (16 / 4);
        scatter_add<16, 16, 32><<<(unsigned)((items + B - 1) / B), B, 0, stream>>>(
            z2, src, dst, agg2, E);
    }
    // 8) final combine -> d_out
    {
        long n = (long)N * 16;
        combine<16, 32><<<(unsigned)((n + B - 1) / B), B, 0, stream>>>(
            z2, agg2, deg, out, N);
    }
}